// DynamicChannelExchangeWithSE_58634893525151
// MI455X (gfx1250) — compile-verified
//
#include <hip/hip_runtime.h>
#include <math.h>

#define HW    (512 * 512)
#define CDIM  128
#define NHALF 64
#define NSPLIT 16
#define SEG   (HW / NSPLIT)        // 16384 elements per split
#define CHW   ((size_t)CDIM * HW)
#define XSTRIDE 132                 // 64-channel LDS tile row stride (pad: 132 % 64 != 0)

typedef __attribute__((ext_vector_type(2))) float v2f;
typedef __attribute__((ext_vector_type(8))) float v8f;

// ---------------------------------------------------------------------------
// Pass 1: per-channel partial sums of lst/gui (split-K, deterministic).
// grid = 2 tensors * 128 channels * 16 splits = 4096 blocks, 256 threads.
// ---------------------------------------------------------------------------
__global__ __launch_bounds__(256) void pass1_reduce(const float* __restrict__ lst,
                                                    const float* __restrict__ gui,
                                                    float* __restrict__ partials) {
    __shared__ float red[256];
    const int bx = blockIdx.x;
    const int t  = bx >> 11;          // 0 = lst, 1 = gui
    const int c  = (bx >> 4) & 127;   // channel
    const int s  = bx & 15;           // split
    const float* src = (t ? gui : lst) + (size_t)c * HW + (size_t)s * SEG;
    const float4* src4 = (const float4*)src;
    const int tid = threadIdx.x;

    float sum = 0.f;
#pragma unroll
    for (int i = 0; i < SEG / 4 / 256; ++i) {   // 16 float4 per thread
        float4 v = src4[tid + i * 256];
        sum += v.x + v.y + v.z + v.w;
    }
    red[tid] = sum;
    __syncthreads();
    for (int off = 128; off > 0; off >>= 1) {
        if (tid < off) red[tid] += red[tid + off];
        __syncthreads();
    }
    if (tid == 0) partials[bx] = red[0];
}

// ---------------------------------------------------------------------------
// Pass 2: finish means, FCNet + SE MLPs, gate m, top-64 selection (ascending).
// Single block of 256 threads.
// ---------------------------------------------------------------------------
__global__ __launch_bounds__(256) void pass2_mask(
    const float* __restrict__ partials, const float* __restrict__ maskin,
    const float* __restrict__ w1, const float* __restrict__ b1,
    const float* __restrict__ w2, const float* __restrict__ b2,
    const float* __restrict__ sew1, const float* __restrict__ seb1,
    const float* __restrict__ sew2, const float* __restrict__ seb2,
    float* __restrict__ m_out, int* __restrict__ sel, int* __restrict__ unsel) {
    __shared__ float pooled[256];    // [lst means 128 | gui means 128]
    __shared__ float h1[128];
    __shared__ float h2[16];
    __shared__ float ms[128];
    __shared__ int   flag[128];
    const int tid = threadIdx.x;

    {   // finish split reduction -> channel means
        float s = 0.f;
#pragma unroll
        for (int i = 0; i < NSPLIT; ++i) s += partials[tid * NSPLIT + i];
        pooled[tid] = s * (1.0f / (float)HW);
    }
    __syncthreads();

    if (tid < 128) {                 // FCNet layer 1: mask[64] @ w1[64,128] + b1, relu
        float s = b1[tid];
        for (int i = 0; i < 64; ++i) s += maskin[i] * w1[i * 128 + tid];
        h1[tid] = fmaxf(s, 0.f);
    }
    if (tid < 16) {                  // SE layer 1: pooled[256] @ se_w1[256,16] + se_b1, relu
        float s = seb1[tid];
        for (int i = 0; i < 256; ++i) s += pooled[i] * sew1[i * 16 + tid];
        h2[tid] = fmaxf(s, 0.f);
    }
    __syncthreads();

    if (tid < 128) {
        float s1 = b2[tid];
        for (int i = 0; i < 128; ++i) s1 += h1[i] * w2[i * 128 + tid];
        float m1 = 1.f / (1.f + __expf(-s1));
        float s2 = seb2[tid];
        for (int i = 0; i < 16; ++i) s2 += h2[i] * sew2[i * 128 + tid];
        float m2 = 1.f / (1.f + __expf(-s2));
        float mv = m1 * m2;
        ms[tid] = mv;
        m_out[tid] = mv;             // third output of the reference
    }
    __syncthreads();

    if (tid < 128) {                 // rank with jax top_k tie-break (lower index wins)
        float mv = ms[tid];
        int rank = 0;
        for (int c = 0; c < 128; ++c) {
            float o = ms[c];
            rank += (o > mv) || (o == mv && c < tid);
        }
        flag[tid] = (rank < NHALF) ? 1 : 0;
    }
    __syncthreads();

    if (tid < 128) {                 // compact to ascending sel[] / unsel[]
        int pos = 0;
        for (int c = 0; c < tid; ++c) pos += flag[c];
        if (flag[tid]) sel[pos] = tid;
        else           unsel[tid - pos] = tid;
    }
}

// ---------------------------------------------------------------------------
// Pass 3: channel exchange. Per WG: 128 pixels. Unselected channels pass
// through; selected channels: D[64,128px] = W[64,64] x Xgather[64,128px]
// via V_WMMA_F32_16X16X4_F32 (each wave owns one 16-pixel tile, 4 M-tiles).
// grid = HW/128 = 2048 blocks, 256 threads (8 wave32).
// ---------------------------------------------------------------------------
__global__ __launch_bounds__(256) void pass3_exchange(
    const float* __restrict__ lst, const float* __restrict__ gui,
    const float* __restrict__ c1w, const float* __restrict__ c1b,
    const float* __restrict__ c2w, const float* __restrict__ c2b,
    const int* __restrict__ selg, const int* __restrict__ unselg,
    float* __restrict__ out_lst, float* __restrict__ out_gui) {
    __shared__ float Xs[NHALF * XSTRIDE];   // 33792 B gathered tile
    __shared__ int   sSel[NHALF];
    __shared__ int   sUnsel[NHALF];

    const int tid = threadIdx.x;
    const int p0  = blockIdx.x * 128;
    if (tid < 64)            sSel[tid]        = selg[tid];
    else if (tid < 128)      sUnsel[tid - 64] = unselg[tid - 64];
    __syncthreads();

    // ---- passthrough: 64 unselected channels, both tensors, float4 ----
    for (int v = tid; v < 2048; v += 256) {
        int row = v >> 5;
        int col = (v & 31) << 2;
        size_t off = (size_t)sUnsel[row] * HW + p0 + col;
        *(float4*)(out_lst + off) = *(const float4*)(lst + off);
        *(float4*)(out_gui + off) = *(const float4*)(gui + off);
    }

    const int lane  = tid & 31;
    const int wv    = tid >> 5;
    const int n0    = wv * 16;          // this wave's pixel sub-tile
    const int row16 = lane & 15;
    const int khalf = lane >> 4;        // 0: K {0,1}, 1: K {2,3} within a step

    // ================= phase A: out_lst[sel] = conv2_w @ gui[sel] ==========
    for (int v = tid; v < 2048; v += 256) {
        int row = v >> 5;
        int col = (v & 31) << 2;
        *(float4*)(&Xs[row * XSTRIDE + col]) =
            *(const float4*)(gui + (size_t)sSel[row] * HW + p0 + col);
    }
    __syncthreads();
    {
        v8f acc[4] = {};
        for (int k = 0; k < 16; ++k) {
            const int kA = k * 4 + khalf * 2;
            v2f b;                       // B 4x16 frag: rows kA, kA+1, col = pixel
            b.x = Xs[kA * XSTRIDE + n0 + row16];
            b.y = Xs[(kA + 1) * XSTRIDE + n0 + row16];
#pragma unroll
            for (int mt = 0; mt < 4; ++mt) {
                const float* wp = c2w + (mt * 16 + row16) * 64 + kA;
                v2f a; a.x = wp[0]; a.y = wp[1];   // A 16x4 frag
                acc[mt] = __builtin_amdgcn_wmma_f32_16x16x4_f32(
                    false, a, false, b, (short)0, acc[mt], false, false);
            }
        }
#pragma unroll
        for (int mt = 0; mt < 4; ++mt)
#pragma unroll
            for (int r = 0; r < 8; ++r) {
                int mrow = mt * 16 + khalf * 8 + r;   // C/D layout: VGPR r -> M=r / r+8
                out_lst[(size_t)sSel[mrow] * HW + p0 + n0 + row16] =
                    acc[mt][r] + c2b[mrow];
            }
    }
    __syncthreads();

    // ================= phase B: out_gui[sel] = conv1_w @ lst[sel] ==========
    for (int v = tid; v < 2048; v += 256) {
        int row = v >> 5;
        int col = (v & 31) << 2;
        *(float4*)(&Xs[row * XSTRIDE + col]) =
            *(const float4*)(lst + (size_t)sSel[row] * HW + p0 + col);
    }
    __syncthreads();
    {
        v8f acc[4] = {};
        for (int k = 0; k < 16; ++k) {
            const int kA = k * 4 + khalf * 2;
            v2f b;
            b.x = Xs[kA * XSTRIDE + n0 + row16];
            b.y = Xs[(kA + 1) * XSTRIDE + n0 + row16];
#pragma unroll
            for (int mt = 0; mt < 4; ++mt) {
                const float* wp = c1w + (mt * 16 + row16) * 64 + kA;
                v2f a; a.x = wp[0]; a.y = wp[1];
                acc[mt] = __builtin_amdgcn_wmma_f32_16x16x4_f32(
                    false, a, false, b, (short)0, acc[mt], false, false);
            }
        }
#pragma unroll
        for (int mt = 0; mt < 4; ++mt)
#pragma unroll
            for (int r = 0; r < 8; ++r) {
                int mrow = mt * 16 + khalf * 8 + r;
                out_gui[(size_t)sSel[mrow] * HW + p0 + n0 + row16] =
                    acc[mt][r] + c1b[mrow];
            }
    }
}

// ---------------------------------------------------------------------------
extern "C" void kernel_launch(void* const* d_in, const int* in_sizes, int n_in,
                              void* d_out, int out_size, void* d_ws, size_t ws_size,
                              hipStream_t stream) {
    const float* lst    = (const float*)d_in[0];
    const float* gui    = (const float*)d_in[1];
    const float* maskin = (const float*)d_in[2];
    const float* w1     = (const float*)d_in[3];
    const float* b1     = (const float*)d_in[4];
    const float* w2     = (const float*)d_in[5];
    const float* b2     = (const float*)d_in[6];
    const float* sew1   = (const float*)d_in[7];
    const float* seb1   = (const float*)d_in[8];
    const float* sew2   = (const float*)d_in[9];
    const float* seb2   = (const float*)d_in[10];
    const float* c1w    = (const float*)d_in[11];
    const float* c1b    = (const float*)d_in[12];
    const float* c2w    = (const float*)d_in[13];
    const float* c2b    = (const float*)d_in[14];

    float* out_lst = (float*)d_out;
    float* out_gui = out_lst + CHW;
    float* m_out   = out_gui + CHW;

    float* partials = (float*)d_ws;                                   // 4096 floats
    int*   sel      = (int*)((char*)d_ws + 4096 * sizeof(float));     // 64 ints
    int*   unsel    = sel + NHALF;                                    // 64 ints

    pass1_reduce<<<4096, 256, 0, stream>>>(lst, gui, partials);
    pass2_mask<<<1, 256, 0, stream>>>(partials, maskin, w1, b1, w2, b2,
                                      sew1, seb1, sew2, seb2, m_out, sel, unsel);
    pass3_exchange<<<HW / 128, 256, 0, stream>>>(lst, gui, c1w, c1b, c2w, c2b,
                                                 sel, unsel, out_lst, out_gui);
}